// Quantize_78365973283370
// MI455X (gfx1250) — compile-verified
//
#include <hip/hip_runtime.h>
#include <hip/hip_bf16.h>
#include <math.h>

typedef __attribute__((ext_vector_type(2))) float v2f;
typedef __attribute__((ext_vector_type(8))) float v8f;
typedef __attribute__((ext_vector_type(4))) unsigned int v4u;
typedef __attribute__((ext_vector_type(8))) int v8i;
typedef __attribute__((ext_vector_type(4))) int v4i;

#define DIM       128
#define NEMBED    10000
#define NROWS     8192                    // 8*32*32
#define NSPLIT    5                       // 10000 = 5 * 2000, divides exactly
#define SPLITLEN  (NEMBED / NSPLIT)       // 2000
#define NT        16                      // N columns per chunk
#define NCHUNK    (SPLITLEN / NT)         // 125, exact -> no OOB guards anywhere
#define MAIN_BLOCK 128                    // 4 waves (wave32)
#define ROWS_PER_BLOCK 64                 // 4 waves * 16 rows
#define MBLOCKS   (NROWS / ROWS_PER_BLOCK) // 128

#if defined(__has_builtin)
#if __has_builtin(__builtin_amdgcn_tensor_load_to_lds) && __has_builtin(__builtin_amdgcn_s_wait_tensorcnt)
#define HAVE_TDM 1
#endif
#endif

// ---------------- kernel 0: c2[e] = sum_d cm[d,e]^2 (coalesced over e) ----
__global__ void k_c2(const float* __restrict__ cm, float* __restrict__ c2) {
  int e = blockIdx.x * blockDim.x + threadIdx.x;
  if (e >= NEMBED) return;
  float s = 0.f;
#pragma unroll 8
  for (int d = 0; d < DIM; ++d) {
    float v = cm[(size_t)d * NEMBED + e];
    s += v * v;
  }
  c2[e] = s;
}

#ifdef HAVE_TDM
// Issue one TDM DMA: cluster_mean tile [DIM rows x NT cols] -> LDS [k][n] row-major.
// D# layout per CDNA5 ISA 8.3/8.4 (group0: count/lds/global/type, group1: dims).
__device__ __forceinline__ void tdm_load_tile(const float* gtile, unsigned lds_off) {
  unsigned long long ga = (unsigned long long)(uintptr_t)gtile;
  v4u g0;
  g0[0] = 1u;                                              // count=1 (valid user D#)
  g0[1] = lds_off;                                         // lds_addr (bytes)
  g0[2] = (unsigned)ga;                                    // global_addr[31:0]
  g0[3] = (unsigned)((ga >> 32) & 0x01FFFFFFull)           // global_addr[56:32]
        | (2u << 30);                                      // type = 2 ("image")
  v8i g1;
  g1[0] = 0x00020000;                                      // data_size=2 (4 bytes)
  g1[1] = (int)((NEMBED & 0xFFFF) << 16);                  // tensor_dim0[15:0]
  g1[2] = (int)((NEMBED >> 16) | ((DIM & 0xFFFF) << 16));  // dim0 hi | tensor_dim1 lo
  g1[3] = (int)((DIM >> 16) | (NT << 16));                 // dim1 hi | tile_dim0=16
  g1[4] = DIM;                                             // tile_dim1=128, tile_dim2=0
  g1[5] = NEMBED;                                          // tensor_dim0_stride lo32
  g1[6] = 0;                                               // stride hi | dim1_stride lo
  g1[7] = 0;
  v4i z = {0, 0, 0, 0};
#if defined(__clang_major__) && (__clang_major__ >= 23)
  v8i z8 = {0, 0, 0, 0, 0, 0, 0, 0};
  __builtin_amdgcn_tensor_load_to_lds(g0, g1, z, z, z8, 0);
#else
  __builtin_amdgcn_tensor_load_to_lds(g0, g1, z, z, 0);
#endif
}
#endif

// ---------------- kernel 1: fused f32 WMMA GEMM + running argmax ----------
// LDS tile layout: [k][n] row-major, k=0..127, n=0..15 (exactly TDM write order).
// B fragment (ISA 4x16 f32): lane -> N=lane%16, K=2*(lane/16)+{0,1}:
//   two ds loads 64B apart; banks (16k+n)%64 distinct for all 32 lanes.
__global__ __launch_bounds__(MAIN_BLOCK) void k_argmax(
    const float* __restrict__ samples, const float* __restrict__ cm,
    const float* __restrict__ c2, float* __restrict__ pval,
    int* __restrict__ pidx) {
  __shared__ float buf[2][NT * DIM];      // 2 x 8 KB double buffer

  const int tid  = threadIdx.x;
  const int lane = tid & 31;
  const int wave = tid >> 5;
  const int mblock = blockIdx.x % MBLOCKS;
  const int split  = blockIdx.x / MBLOCKS;
  const int row0 = mblock * ROWS_PER_BLOCK + wave * 16;
  const int hi = lane >> 4;
  const int ln = lane & 15;

  // Preload A fragments (ISA 16x4 f32 A layout: M=lane%16, K=2*(lane/16)+{0,1}).
  v2f afrag[32];
  const float* ap = samples + (size_t)(row0 + ln) * DIM + 2 * hi;
#pragma unroll
  for (int s = 0; s < 32; ++s)
    afrag[s] = *(const v2f*)(ap + s * 4);

  const int e_base = split * SPLITLEN;
  const float* gbase = cm + e_base;       // tile c starts at gbase + c*NT

  float best[8];
  int   bidx[8];
#pragma unroll
  for (int i = 0; i < 8; ++i) { best[i] = -INFINITY; bidx[i] = 0; }

  // ---- prologue: fetch tile 0 into buf[0] ----
#ifdef HAVE_TDM
  if (wave == 0) {
    tdm_load_tile(gbase, (unsigned)(uintptr_t)&buf[0][0]);
    __builtin_amdgcn_s_wait_tensorcnt(0);
  }
#else
  for (int i = tid; i < NT * DIM; i += MAIN_BLOCK) {
    int n = i & (NT - 1), d = i >> 4;
    buf[0][d * NT + n] = cm[(size_t)d * NEMBED + e_base + n];
  }
#endif
  __syncthreads();

  for (int c = 0; c < NCHUNK; ++c) {
    const int cur = c & 1;
    // ---- kick off async fetch of tile c+1 into the other buffer ----
#ifdef HAVE_TDM
    if (wave == 0 && c + 1 < NCHUNK)
      tdm_load_tile(gbase + (c + 1) * NT, (unsigned)(uintptr_t)&buf[cur ^ 1][0]);
#else
    if (c + 1 < NCHUNK)
      for (int i = tid; i < NT * DIM; i += MAIN_BLOCK) {
        int n = i & (NT - 1), d = i >> 4;
        buf[cur ^ 1][d * NT + n] = cm[(size_t)d * NEMBED + e_base + (c + 1) * NT + n];
      }
#endif

    // ---- 32 K-steps of WMMA over tile c, 4 independent accumulators ----
    v8f acc0 = {}, acc1 = {}, acc2 = {}, acc3 = {};
    const float* bbase = &buf[cur][ln];
#pragma unroll
    for (int s = 0; s < 32; s += 4) {
      v2f b0, b1, b2, b3;
      b0.x = bbase[((s + 0) * 4 + 2 * hi) * NT]; b0.y = bbase[((s + 0) * 4 + 2 * hi + 1) * NT];
      b1.x = bbase[((s + 1) * 4 + 2 * hi) * NT]; b1.y = bbase[((s + 1) * 4 + 2 * hi + 1) * NT];
      b2.x = bbase[((s + 2) * 4 + 2 * hi) * NT]; b2.y = bbase[((s + 2) * 4 + 2 * hi + 1) * NT];
      b3.x = bbase[((s + 3) * 4 + 2 * hi) * NT]; b3.y = bbase[((s + 3) * 4 + 2 * hi + 1) * NT];
      acc0 = __builtin_amdgcn_wmma_f32_16x16x4_f32(false, afrag[s + 0], false, b0, (short)0, acc0, false, false);
      acc1 = __builtin_amdgcn_wmma_f32_16x16x4_f32(false, afrag[s + 1], false, b1, (short)0, acc1, false, false);
      acc2 = __builtin_amdgcn_wmma_f32_16x16x4_f32(false, afrag[s + 2], false, b2, (short)0, acc2, false, false);
      acc3 = __builtin_amdgcn_wmma_f32_16x16x4_f32(false, afrag[s + 3], false, b3, (short)0, acc3, false, false);
    }

    // score = c2[e] - 2*dot ; ||s||^2 is row-constant, irrelevant for argmax.
    int e = e_base + c * NT + ln;
    float c2v = c2[e];
#pragma unroll
    for (int i = 0; i < 8; ++i) {
      float dot = acc0[i] + acc1[i] + acc2[i] + acc3[i];
      float val = c2v - 2.f * dot;
      if (val > best[i]) { best[i] = val; bidx[i] = e; }   // strict > keeps first max
    }

#ifdef HAVE_TDM
    if (wave == 0 && c + 1 < NCHUNK)
      __builtin_amdgcn_s_wait_tensorcnt(0);   // DMA done before anyone reads it
#endif
    __syncthreads();
  }

  // Butterfly argmax across the 16 lanes of each half (C layout: lane=N, vgpr=M).
#pragma unroll
  for (int i = 0; i < 8; ++i) {
#pragma unroll
    for (int off = 1; off < 16; off <<= 1) {
      float ov = __shfl_xor(best[i], off, 32);
      int   oi = __shfl_xor(bidx[i], off, 32);
      if (ov > best[i] || (ov == best[i] && oi < bidx[i])) { best[i] = ov; bidx[i] = oi; }
    }
  }
  if (ln == 0) {   // lane 0 -> rows row0+0..7 ; lane 16 -> rows row0+8..15
#pragma unroll
    for (int i = 0; i < 8; ++i) {
      int r = row0 + 8 * hi + i;
      pval[(size_t)r * NSPLIT + split] = best[i];
      pidx[(size_t)r * NSPLIT + split] = bidx[i];
    }
  }
}

// ---------------- kernel 2: fold the NSPLIT partials per row --------------
__global__ void k_reduce(const float* __restrict__ pval, const int* __restrict__ pidx,
                         int* __restrict__ fidx, float* __restrict__ out_idx_f) {
  int r = blockIdx.x * blockDim.x + threadIdx.x;
  if (r >= NROWS) return;
  float bv = -INFINITY; int bi = 0;
#pragma unroll
  for (int s = 0; s < NSPLIT; ++s) {        // ascending split => ascending indices,
    float v = pval[(size_t)r * NSPLIT + s]; //  strict > preserves first-occurrence
    int  ix = pidx[(size_t)r * NSPLIT + s];
    if (v > bv) { bv = v; bi = ix; }
  }
  fidx[r] = bi;
  out_idx_f[r] = (float)bi;
}

// ---------------- kernel 3: gather quantize column + diff -----------------
__global__ __launch_bounds__(128) void k_quant(
    const float* __restrict__ samples, const float* __restrict__ cm,
    const int* __restrict__ fidx, float* __restrict__ out_q,
    float* __restrict__ out_diff) {
  __shared__ float part[4];
  int r = blockIdx.x;
  int d = threadIdx.x;
  int idx = fidx[r];
  float s = samples[(size_t)r * DIM + d];
  float q = cm[(size_t)d * NEMBED + idx];
  out_q[(size_t)r * DIM + d] = q;
  float e = (s - q) * (s - q);
#pragma unroll
  for (int off = 16; off > 0; off >>= 1) e += __shfl_xor(e, off, 32);
  if ((d & 31) == 0) part[d >> 5] = e;
  __syncthreads();
  if (d == 0) out_diff[r] = (part[0] + part[1] + part[2] + part[3]) * (1.f / DIM);
}

extern "C" void kernel_launch(void* const* d_in, const int* in_sizes, int n_in,
                              void* d_out, int out_size, void* d_ws, size_t ws_size,
                              hipStream_t stream) {
  const float* samples = (const float*)d_in[0];   // [8,32,32,128] -> [8192,128]
  const float* cm      = (const float*)d_in[1];   // [128,10000]

  float* out      = (float*)d_out;
  float* out_q    = out;                              // 8192*128
  float* out_idx  = out + (size_t)NROWS * DIM;        // 8192 (indices as float)
  float* out_diff = out_idx + NROWS;                  // 8192

  char*  ws   = (char*)d_ws;
  float* c2   = (float*)ws;                                // 40000 B (pad 40960)
  float* pval = (float*)(ws + 40960);                      // 8192*5*4 = 163840 B
  int*   pidx = (int*)  (ws + 40960 + 163840);             // 163840 B
  int*   fidx = (int*)  (ws + 40960 + 2 * 163840);         // 32768 B

  k_c2    <<<(NEMBED + 255) / 256, 256, 0, stream>>>(cm, c2);
  k_argmax<<<MBLOCKS * NSPLIT, MAIN_BLOCK, 0, stream>>>(samples, cm, c2, pval, pidx);
  k_reduce<<<(NROWS + 255) / 256, 256, 0, stream>>>(pval, pidx, fidx, out_idx);
  k_quant <<<NROWS, 128, 0, stream>>>(samples, cm, fidx, out_q, out_diff);
}